// Fractal2DDiff_66726611911089
// MI455X (gfx1250) — compile-verified
//
#include <hip/hip_runtime.h>
#include <hip/hip_bf16.h>

typedef __attribute__((ext_vector_type(2))) float v2f;
typedef __attribute__((ext_vector_type(8))) float v8f;

#define TFINV 1.0e4f
#define NV 16
#define NC 10

// Hardware V_RCP_F32 (TRANS op, co-executes with WMMA) + one Newton step:
// ~0.5 ulp, avoids the ~10-op IEEE div_scale/div_fixup/div_fmas expansion.
// Caller must keep x finite and >= ~1e-30 (NR step NaNs on inf/0).
__device__ __forceinline__ float fastrcp(float x) {
    float r = __builtin_amdgcn_rcpf(x);
    return r * (2.0f - x * r);
}

// sigmoid(z / 1e-4). expf saturates to inf for z < ~-0.009; clamp so the
// NR-refined rcp stays NaN-free (yields 1e-30 instead of exactly 0).
__device__ __forceinline__ float sigT(float z) {
    float e = __expf(-z * TFINV);
    e = fminf(e, 1.0e30f);
    return fastrcp(1.0f + e);
}

// swap lane p <-> lane p+16 within the wave32 (ds_swizzle group-of-32:
// xor=0x10, or=0, and=0x1f -> offset 0x401F)
__device__ __forceinline__ float swap16(float v) {
    return __int_as_float(__builtin_amdgcn_ds_swizzle(__float_as_int(v), 0x401f));
}

// ---------------------------------------------------------------------------
// Precompute (one tiny block): ws layout in floats:
//   [0,16)    sp  = clip(split_points, 0.1, 0.9)
//   [16,32)   d   = sigmoid(split_directions_logits)
//   [32,288)  Cl  = softmax(csl[:,0,:]/TF)   row-major [16][16]
//   [288,544) Cr  = softmax(csl[:,1,:]/TF)   row-major [16][16]
//   [544,704) cls = softmax(class_logits/TF) row-major [16][10]
// ---------------------------------------------------------------------------
__global__ void fract_precompute(const float* __restrict__ sp_in,
                                 const float* __restrict__ dlog,
                                 const float* __restrict__ clslog,
                                 const float* __restrict__ csl,
                                 float* __restrict__ ws) {
    int t = threadIdx.x;
    if (t >= NV) return;
    ws[t] = fminf(fmaxf(sp_in[t], 0.1f), 0.9f);
    {
        float e = fminf(__expf(-dlog[t]), 1.0e30f);
        ws[NV + t] = fastrcp(1.0f + e);
    }
    // Cl row t
    {
        const float* row = csl + t * 2 * NV;
        float m = row[0];
        for (int i = 1; i < NV; i++) m = fmaxf(m, row[i]);
        float e[NV], s = 0.0f;
        for (int i = 0; i < NV; i++) { e[i] = __expf((row[i] - m) * TFINV); s += e[i]; }
        float is = fastrcp(s);
        for (int i = 0; i < NV; i++) ws[32 + t * NV + i] = e[i] * is;
    }
    // Cr row t
    {
        const float* row = csl + t * 2 * NV + NV;
        float m = row[0];
        for (int i = 1; i < NV; i++) m = fmaxf(m, row[i]);
        float e[NV], s = 0.0f;
        for (int i = 0; i < NV; i++) { e[i] = __expf((row[i] - m) * TFINV); s += e[i]; }
        float is = fastrcp(s);
        for (int i = 0; i < NV; i++) ws[288 + t * NV + i] = e[i] * is;
    }
    // cls row t
    {
        const float* row = clslog + t * NC;
        float m = row[0];
        for (int i = 1; i < NC; i++) m = fmaxf(m, row[i]);
        float e[NC], s = 0.0f;
        for (int i = 0; i < NC; i++) { e[i] = __expf((row[i] - m) * TFINV); s += e[i]; }
        float is = fastrcp(s);
        for (int i = 0; i < NC; i++) ws[544 + t * NC + i] = e[i] * is;
    }
}

// ---------------------------------------------------------------------------
// Main kernel: one wave32 per tile of 16 points.
// Lanes p and p+16 both own point p; lo half handles B-slot nodes
// {0,1,4,5,8,9,12,13}, hi half {2,3,6,7,10,11,14,15}  (slot j -> node
// 4*(j/2) + (j%2) + 2*hi), which is exactly WMMA f32 16x16x4 B-layout for
// the K=16 chain. D = next_probs^T (lane=point, nodes 0-7 lo / 8-15 hi).
// ---------------------------------------------------------------------------
__global__ __launch_bounds__(256) void fract_main(const float* __restrict__ xs,
                                                  const float* __restrict__ ys,
                                                  const float* __restrict__ ws,
                                                  const int* __restrict__ mdp,
                                                  float* __restrict__ out,
                                                  int nTiles) {
    const int wave = threadIdx.x >> 5;
    const int lane = threadIdx.x & 31;
    const int tile = blockIdx.x * 8 + wave;
    if (tile >= nTiles) return;   // whole wave uniform -> EXEC all-1s inside

    const int p    = lane & 15;
    const int hi   = lane >> 4;
    const int base = tile * 16;
    const int md   = *mdp;

    const float* wsp  = ws;
    const float* wd   = ws + 16;
    const float* wCl  = ws + 32;
    const float* wCr  = ws + 288;
    const float* wcls = ws + 544;

    const float px = xs[base + p];
    const float py = ys[base + p];
    float mnx = 0.0f, mxx = 1.0f, mny = 0.0f, mxy = 1.0f;

    // A-layout (lane = out-node m; hi half carries K+2 per WMMA A spec):
    // acl[j] = Cl^T[m][k(j,hi)] = Cl[k][m]
    float acl[8], acr[8], sps[8], ds[8];
#pragma unroll
    for (int j = 0; j < 8; j++) {
        const int nk = 4 * (j >> 1) + (j & 1) + 2 * hi;
        acl[j] = wCl[nk * 16 + p];
        acr[j] = wCr[nk * 16 + p];
        sps[j] = wsp[nk];
        ds[j]  = wd[nk];
    }

    // node_probs in B-layout; node 0 == slot 0 of lo half
    float probsB[8];
#pragma unroll
    for (int j = 0; j < 8; j++) probsB[j] = 0.0f;
    if (!hi) probsB[0] = 1.0f;

    for (int it = 0; it < md; ++it) {
        float leftB[8], rightB[8];
        float sumL = 0.0f, sumR = 0.0f;
        float aMnX = 0.0f, aMxX = 0.0f, aMnY = 0.0f, aMxY = 0.0f;
#pragma unroll
        for (int j = 0; j < 8; j++) {
            const float spj = sps[j], dj = ds[j], oj = 1.0f - dj;
            const float sx = mnx + spj * (mxx - mnx);
            const float sy = mny + spj * (mxy - mny);
            const float pl = oj * sigT(sx - px) + dj * sigT(sy - py);
            const float l = pl * probsB[j];
            const float r = probsB[j] - l;
            leftB[j] = l;
            rightB[j] = r;
            sumL += l;
            sumR += r;
            aMxX += (oj * sx + dj * mxx) * l;   // lb_max_x * left
            aMnX += (oj * sx + dj * mnx) * r;   // rb_min_x * right
            aMxY += (dj * sy + oj * mxy) * l;   // lb_max_y * left
            aMnY += (dj * sy + oj * mny) * r;   // rb_min_y * right
        }
        // combine the two 8-node halves per point
        sumL += swap16(sumL);
        sumR += swap16(sumR);
        aMnX += swap16(aMnX);
        aMxX += swap16(aMxX);
        aMnY += swap16(aMnY);
        aMxY += swap16(aMxY);
        const float inv = fastrcp(fmaxf(sumL + sumR, 1e-10f));
        mnx = (mnx * sumL + aMnX) * inv;
        mxx = (aMxX + mxx * sumR) * inv;
        mny = (mny * sumL + aMnY) * inv;
        mxy = (aMxY + mxy * sumR) * inv;

        // next_probs^T = Cl^T @ left^T + Cr^T @ right^T  (K=16 as 4+4 WMMAs)
        v8f D = {};
#pragma unroll
        for (int i = 0; i < 4; i++) {
            v2f A, Bv;
            A.x = acl[2 * i]; A.y = acl[2 * i + 1];
            Bv.x = leftB[2 * i]; Bv.y = leftB[2 * i + 1];
            D = __builtin_amdgcn_wmma_f32_16x16x4_f32(false, A, false, Bv,
                                                      (short)0, D, false, false);
        }
#pragma unroll
        for (int i = 0; i < 4; i++) {
            v2f A, Bv;
            A.x = acr[2 * i]; A.y = acr[2 * i + 1];
            Bv.x = rightB[2 * i]; Bv.y = rightB[2 * i + 1];
            D = __builtin_amdgcn_wmma_f32_16x16x4_f32(false, A, false, Bv,
                                                      (short)0, D, false, false);
        }

        // normalize per point (sum over all 16 nodes)
        float s = D[0] + D[1] + D[2] + D[3] + D[4] + D[5] + D[6] + D[7];
        s += swap16(s);
        const float invp = fastrcp(fmaxf(s, 1e-10f));

        // D-layout (nodes 0-7 lo / 8-15 hi) -> B-layout via half-lane swap
        const int m0[4] = {0, 1, 4, 5};
        const int m1[4] = {2, 3, 6, 7};
#pragma unroll
        for (int j = 0; j < 8; j++) {
            const int q = j & 3;
            const float send = (j >= 4) ? D[m0[q]] : D[m1[q]];
            const float t = swap16(send);
            const float keep = (j < 4) ? D[m0[q]] : D[m1[q]];
            const float val = (j < 4) ? (hi ? t : keep) : (hi ? keep : t);
            probsB[j] = val * invp;
        }
    }

    // class_probs = node_probs @ cls  (C=10, tiny tail)
#pragma unroll
    for (int c = 0; c < NC; c++) {
        float acc = 0.0f;
#pragma unroll
        for (int j = 0; j < 8; j++) {
            const int nv = 4 * (j >> 1) + (j & 1) + 2 * hi;
            acc += probsB[j] * wcls[nv * NC + c];
        }
        acc += swap16(acc);
        if (!hi) out[(size_t)(base + p) * NC + c] = acc;
    }
}

extern "C" void kernel_launch(void* const* d_in, const int* in_sizes, int n_in,
                              void* d_out, int out_size, void* d_ws, size_t ws_size,
                              hipStream_t stream) {
    const float* sp  = (const float*)d_in[0];
    const float* dl  = (const float*)d_in[1];
    const float* cls = (const float*)d_in[2];
    const float* csl = (const float*)d_in[3];
    const float* xs  = (const float*)d_in[4];
    const float* ys  = (const float*)d_in[5];
    const int*   md  = (const int*)d_in[6];
    float* ws  = (float*)d_ws;
    float* out = (float*)d_out;

    const int B = in_sizes[4];
    const int nTiles = B / 16;            // 1024*1024/16 = 65536
    fract_precompute<<<1, 32, 0, stream>>>(sp, dl, cls, csl, ws);
    const int blocks = (nTiles + 7) / 8;  // 8 waves (tiles) per 256-thread block
    fract_main<<<blocks, 256, 0, stream>>>(xs, ys, ws, md, out, nTiles);
}